// GraphEncoder_26860725469213
// MI455X (gfx1250) — compile-verified
//
#include <hip/hip_runtime.h>

#define N_NODES 50000

typedef __attribute__((ext_vector_type(2))) float v2f;
typedef __attribute__((ext_vector_type(8))) float v8f;

// ---------------------------------------------------------------- utilities
__global__ void zero_kernel(float* __restrict__ p, size_t n) {
    size_t i = (size_t)blockIdx.x * blockDim.x + threadIdx.x;
    size_t stride = (size_t)gridDim.x * blockDim.x;
    for (; i < n; i += stride) p[i] = 0.0f;
}

// degree[dst] += 1 for every edge (fp32 atomics; reused across all 4 layers)
__global__ void degree_kernel(const long long* __restrict__ ei,
                              float* __restrict__ cnt, int E) {
    int e = blockIdx.x * blockDim.x + threadIdx.x;
    if (e < E) {
        int dst = (int)ei[(size_t)E + e];
        atomicAdd(cnt + dst, 1.0f);
    }
}

__global__ void invert_kernel(float* __restrict__ cnt, int n) {
    int i = blockIdx.x * blockDim.x + threadIdx.x;
    if (i < n) cnt[i] = 1.0f / fmaxf(cnt[i], 1.0f);
}

// scatter-add: s[dst][f..f+3] += in[src][f..f+3]; one float4 group per thread
__global__ void scatter_kernel(const float* __restrict__ in,
                               const long long* __restrict__ ei,
                               float* __restrict__ s, int E, int d) {
    int idx = blockIdx.x * blockDim.x + threadIdx.x;
    int g = d >> 2;                      // float4 groups per row
    int e = idx / g;
    if (e >= E) return;
    int f = (idx - e * g) << 2;
    int src = (int)ei[e];
    int dst = (int)ei[(size_t)E + e];
    const float4 v = *(const float4*)(in + (size_t)src * d + f);
    float* o = s + (size_t)dst * d + f;
    atomicAdd(o + 0, v.x);
    atomicAdd(o + 1, v.y);
    atomicAdd(o + 2, v.z);
    atomicAdd(o + 3, v.w);
}

// ------------------------------------------------- fused SAGE linear (WMMA)
// out = act( (S * inv) @ Wl^T + bl + X @ Wr^T )
// V_WMMA_F32_16X16X4_F32, wave32. Each wave: one 16-wide N tile x BM M-tiles.
// A (16x4 f32): lanes 0-15 hold K=k0,k0+1 (VGPR0,1); lanes 16-31 hold K=k0+2,k0+3.
// B (4x16 f32): same K split across lane halves; column = tileN + (lane&15).
// C/D (16x16 f32, 8 VGPRs): row = tileM + i + (lane>=16 ? 8 : 0), col = tileN+(lane&15).
template <int D_IN, int H, bool RELU, int BM>
__global__ __launch_bounds__(256) void sage_gemm_kernel(
    const float* __restrict__ S, const float* __restrict__ X,
    const float* __restrict__ inv, const float* __restrict__ Wl,
    const float* __restrict__ bl, const float* __restrict__ Wr,
    float* __restrict__ out) {
    constexpr int MT = (N_NODES + 15) / 16;   // 3125 M tiles
    constexpr int MG = (MT + BM - 1) / BM;    // M tile groups per wave
    constexpr int NT = H / 16;

    const int wave = (blockIdx.x * blockDim.x + threadIdx.x) >> 5;
    const int lane = threadIdx.x & 31;
    if (wave >= MG * NT) return;              // wave-uniform: EXEC stays all-1s

    const int tileN = (wave % NT) * 16;
    const int mg    = wave / NT;

    const int lm = lane & 15;
    const int kh = (lane >> 4) << 1;          // 0 or 2: K sub-pair for this half

    // B fragments: column (tileN+lm) of Wl/Wr (row-major [H, D_IN] => W[n][k])
    const float* wl = Wl + (size_t)(tileN + lm) * D_IN + kh;
    const float* wr = Wr + (size_t)(tileN + lm) * D_IN + kh;

    int   rows[BM];
    float invm[BM];
#pragma unroll
    for (int mt = 0; mt < BM; ++mt) {
        int r = mg * (BM * 16) + mt * 16 + lm;
        rows[mt] = (r < N_NODES) ? r : (N_NODES - 1);   // clamp (stores guarded)
        invm[mt] = inv[rows[mt]];
    }

    v8f c[BM] = {};

    // phase 1: (S * inv) @ Wl^T  — one B load reused by BM independent WMMAs
    for (int k0 = 0; k0 < D_IN; k0 += 4) {
        v2f b = *(const v2f*)(wl + k0);
#pragma unroll
        for (int mt = 0; mt < BM; ++mt) {
            v2f a = *(const v2f*)(S + (size_t)rows[mt] * D_IN + k0 + kh);
            a *= invm[mt];
            c[mt] = __builtin_amdgcn_wmma_f32_16x16x4_f32(
                false, a, false, b, (short)0, c[mt], false, false);
        }
    }
    // phase 2: X @ Wr^T — accumulate into the same tiles
    for (int k0 = 0; k0 < D_IN; k0 += 4) {
        v2f b = *(const v2f*)(wr + k0);
#pragma unroll
        for (int mt = 0; mt < BM; ++mt) {
            v2f a = *(const v2f*)(X + (size_t)rows[mt] * D_IN + k0 + kh);
            c[mt] = __builtin_amdgcn_wmma_f32_16x16x4_f32(
                false, a, false, b, (short)0, c[mt], false, false);
        }
    }

    const int col = tileN + lm;
    const float bias = bl[col];
#pragma unroll
    for (int mt = 0; mt < BM; ++mt) {
        const int rbase = mg * (BM * 16) + mt * 16 + ((lane >> 4) << 3);
#pragma unroll
        for (int i = 0; i < 8; ++i) {
            int r = rbase + i;
            if (r < N_NODES) {
                float v = c[mt][i] + bias;
                if (RELU) v = fmaxf(v, 0.0f);
                out[(size_t)r * H + col] = v;
            }
        }
    }
}

template <int D_IN, int H, bool RELU>
static void launch_gemm(const float* S, const float* X, const float* inv,
                        const float* Wl, const float* bl, const float* Wr,
                        float* out, hipStream_t stream) {
    constexpr int BM = 4;
    constexpr int MT = (N_NODES + 15) / 16;
    constexpr int MG = (MT + BM - 1) / BM;
    constexpr int NT = H / 16;
    const int nthreads = MG * NT * 32;
    const int nblocks = (nthreads + 255) / 256;
    sage_gemm_kernel<D_IN, H, RELU, BM>
        <<<nblocks, 256, 0, stream>>>(S, X, inv, Wl, bl, Wr, out);
}

// ------------------------------------------------------------------ driver
extern "C" void kernel_launch(void* const* d_in, const int* in_sizes, int n_in,
                              void* d_out, int out_size, void* d_ws, size_t ws_size,
                              hipStream_t stream) {
    const float*     x   = (const float*)d_in[0];
    const long long* ei  = (const long long*)d_in[1];   // int64 [2, E]
    const float* Wl1 = (const float*)d_in[2];
    const float* bl1 = (const float*)d_in[3];
    const float* Wr1 = (const float*)d_in[4];
    const float* Wl2 = (const float*)d_in[5];
    const float* bl2 = (const float*)d_in[6];
    const float* Wr2 = (const float*)d_in[7];
    const float* Wl3 = (const float*)d_in[8];
    const float* bl3 = (const float*)d_in[9];
    const float* Wr3 = (const float*)d_in[10];
    const float* Wl4 = (const float*)d_in[11];
    const float* bl4 = (const float*)d_in[12];
    const float* Wr4 = (const float*)d_in[13];
    const int E = in_sizes[1] / 2;

    float* ws  = (float*)d_ws;
    float* inv = ws;                              // N
    float* s   = inv + N_NODES;                   // N*128 (per-layer scratch)
    float* h1  = s  + (size_t)N_NODES * 128;      // N*64
    float* h2  = h1 + (size_t)N_NODES * 64;       // N*128
    float* h3  = h2 + (size_t)N_NODES * 128;      // N*128
    float* out = (float*)d_out;                   // N*64

    // reciprocal mean-degree, shared by all 4 layers
    zero_kernel<<<256, 256, 0, stream>>>(inv, (size_t)N_NODES);
    degree_kernel<<<(E + 255) / 256, 256, 0, stream>>>(ei, inv, E);
    invert_kernel<<<(N_NODES + 255) / 256, 256, 0, stream>>>(inv, N_NODES);

    // layer 1: x (128) -> h1 (64), relu
    zero_kernel<<<2048, 256, 0, stream>>>(s, (size_t)N_NODES * 128);
    scatter_kernel<<<(E * 32 + 255) / 256, 256, 0, stream>>>(x, ei, s, E, 128);
    launch_gemm<128, 64, true>(s, x, inv, Wl1, bl1, Wr1, h1, stream);

    // layer 2: h1 (64) -> h2 (128), relu
    zero_kernel<<<2048, 256, 0, stream>>>(s, (size_t)N_NODES * 64);
    scatter_kernel<<<(E * 16 + 255) / 256, 256, 0, stream>>>(h1, ei, s, E, 64);
    launch_gemm<64, 128, true>(s, h1, inv, Wl2, bl2, Wr2, h2, stream);

    // layer 3: h2 (128) -> h3 (128), relu
    zero_kernel<<<2048, 256, 0, stream>>>(s, (size_t)N_NODES * 128);
    scatter_kernel<<<(E * 32 + 255) / 256, 256, 0, stream>>>(h2, ei, s, E, 128);
    launch_gemm<128, 128, true>(s, h2, inv, Wl3, bl3, Wr3, h3, stream);

    // layer 4: h3 (128) -> out (64), no relu
    zero_kernel<<<2048, 256, 0, stream>>>(s, (size_t)N_NODES * 128);
    scatter_kernel<<<(E * 32 + 255) / 256, 256, 0, stream>>>(h3, ei, s, E, 128);
    launch_gemm<128, 64, false>(s, h3, inv, Wl4, bl4, Wr4, out, stream);
}